// LogLikelihoodLayer_54030688583920
// MI455X (gfx1250) — compile-verified
//
#include <hip/hip_runtime.h>
#include <hip/hip_bf16.h>
#include <math.h>

typedef __attribute__((ext_vector_type(2))) float v2f;
typedef __attribute__((ext_vector_type(8))) float v8f;

#define KSTATES 12
#define CDIM 64
#define PTS_PER_WG 16
#define PITCH 65                  // LDS row pitch (floats): 65 mod 64 banks -> conflict-free column walks
#define PT_STRIDE (CDIM * PITCH)  // 4160 floats per point matrix
#define NTILES 256                // 4096 flattened (i,j) columns / 16
#define WAVES 8
#define BLOCK 256

__global__ __launch_bounds__(BLOCK) void gmm_nll_kernel(
    const float* __restrict__ y,      // [P,64]
    const float* __restrict__ theta,  // [P,12]
    const float* __restrict__ means,  // [12,64]
    const float* __restrict__ covs,   // [12,64,64] (flattened 12 x 4096)
    float* __restrict__ partial)      // [gridDim.x]
{
  extern __shared__ float lds[];
  float* covS   = lds;                              // 16 * 4160
  float* diffS  = covS + PTS_PER_WG * PT_STRIDE;    // 16 * 64
  float* alphaS = diffS + PTS_PER_WG * CDIM;        // 16 * 16
  float* redS   = alphaS + PTS_PER_WG * 16;         // 8

  const int tid  = threadIdx.x;
  const int lane = tid & 31;
  const int wave = tid >> 5;
  const int p0   = blockIdx.x * PTS_PER_WG;

  // ---------- Phase A: softmax(theta) -> alpha ; diff = y - alpha@means ----------
  if (tid < PTS_PER_WG) {
    const float* th = theta + (size_t)(p0 + tid) * KSTATES;
    float t[KSTATES];
    float mx = -1e30f;
    #pragma unroll
    for (int k = 0; k < KSTATES; ++k) { t[k] = th[k]; mx = fmaxf(mx, t[k]); }
    float s = 0.f;
    #pragma unroll
    for (int k = 0; k < KSTATES; ++k) { float e = __expf(t[k] - mx); alphaS[tid * 16 + k] = e; s += e; }
    float inv = 1.f / s;
    #pragma unroll
    for (int k = 0; k < KSTATES; ++k) alphaS[tid * 16 + k] *= inv;
  }
  __syncthreads();

  for (int e = tid; e < PTS_PER_WG * CDIM; e += BLOCK) {
    int p = e >> 6, c = e & 63;
    float s = y[(size_t)(p0 + p) * CDIM + c];
    #pragma unroll
    for (int k = 0; k < KSTATES; ++k) s -= alphaS[p * 16 + k] * means[k * CDIM + c];
    diffS[e] = s;
  }
  __syncthreads();

  // ---------- Phase B: cov[p] = sum_k alpha[p][k] * covs[k]  via V_WMMA_F32_16X16X4_F32 ----
  // GEMM: [16 points x 12(K)] x [12 x 4096 flattened (i,j)], K in 3 chunks of 4.
  {
    const int m     = lane & 15;  // point row within tile / N column selector
    const int khalf = lane >> 4;  // A: K pair {0,1} vs {2,3}; D: M rows 0-7 vs 8-15
    for (int tl = wave; tl < NTILES; tl += WAVES) {
      int ng = tl * 16 + m;       // flattened (i,j) column index
      int ii = ng >> 6, jj = ng & 63;
      v8f acc = {};
      #pragma unroll
      for (int kc = 0; kc < 3; ++kc) {
        int kk = kc * 4 + 2 * khalf;
        v2f a, b;
        a[0] = alphaS[m * 16 + kk];
        a[1] = alphaS[m * 16 + kk + 1];
        b[0] = covs[(size_t)kk * 4096 + ng];
        b[1] = covs[(size_t)(kk + 1) * 4096 + ng];
        acc = __builtin_amdgcn_wmma_f32_16x16x4_f32(false, a, false, b,
                                                    (short)0, acc, false, false);
      }
      float eps = (ii == jj) ? 1e-8f : 0.f;
      int mbase = 8 * khalf;
      #pragma unroll
      for (int v = 0; v < 8; ++v)
        covS[(size_t)(mbase + v) * PT_STRIDE + ii * PITCH + jj] = acc[v] + eps;
    }
  }
  __syncthreads();

  // ---------- Phase C: per-wave Cholesky + forward solve for 2 points ----------
  float res = 0.f;
  for (int pp = 0; pp < 2; ++pp) {
    float* covp = covS + (size_t)(wave * 2 + pp) * PT_STRIDE;
    float* dfp  = diffS + (wave * 2 + pp) * CDIM;
    float logsum = 0.f, quad = 0.f;

    for (int j = 0; j < CDIM; ++j) {
      __syncthreads();  // trailing updates of step j-1 visible
      float d   = sqrtf(covp[j * PITCH + j]);   // broadcast LDS read
      float inv = 1.f / d;
      logsum += __logf(d);
      for (int i = j + lane; i < CDIM; i += 32)
        covp[i * PITCH + j] *= inv;             // column j incl. diag -> L[:,j]
      __syncthreads();
      for (int i = j + 1 + lane; i < CDIM; i += 32) {
        float Lij = covp[i * PITCH + j];
        for (int q = j + 1; q <= i; ++q)        // lower-triangular rank-1 update
          covp[i * PITCH + q] -= Lij * covp[q * PITCH + j];
      }
    }
    __syncthreads();

    for (int j = 0; j < CDIM; ++j) {            // z = L^{-1} diff
      float zj = dfp[j] / covp[j * PITCH + j];  // broadcast reads
      quad += zj * zj;
      for (int i = j + 1 + lane; i < CDIM; i += 32)
        dfp[i] -= covp[i * PITCH + j] * zj;
      __syncthreads();
    }
    // 0.5*(logdet + quad) = sum(log Ljj) + 0.5*quad
    res += logsum + 0.5f * quad;
  }

  if (lane == 0) redS[wave] = res;
  __syncthreads();
  if (tid == 0) {
    float s = 0.f;
    #pragma unroll
    for (int w = 0; w < WAVES; ++w) s += redS[w];
    partial[blockIdx.x] = s;
  }
}

__global__ __launch_bounds__(BLOCK) void reduce_kernel(const float* __restrict__ in,
                                                       float* __restrict__ out, int n) {
  __shared__ float s[BLOCK];
  float acc = 0.f;
  for (int i = threadIdx.x; i < n; i += BLOCK) acc += in[i];
  s[threadIdx.x] = acc;
  __syncthreads();
  for (int o = BLOCK / 2; o > 0; o >>= 1) {
    if ((int)threadIdx.x < o) s[threadIdx.x] += s[threadIdx.x + o];
    __syncthreads();
  }
  if (threadIdx.x == 0) out[0] = s[0];
}

extern "C" void kernel_launch(void* const* d_in, const int* in_sizes, int n_in,
                              void* d_out, int out_size, void* d_ws, size_t ws_size,
                              hipStream_t stream) {
  const float* y     = (const float*)d_in[0];  // y_portioned [B,T,C]
  const float* theta = (const float*)d_in[1];  // theta_ast   [B,T,K]
  const float* means = (const float*)d_in[2];  // mean_matrix [K,C]
  const float* covs  = (const float*)d_in[3];  // covariance_matrix [K,C,C]
  float* out     = (float*)d_out;
  float* partial = (float*)d_ws;

  const int P   = in_sizes[1] / KSTATES;   // 16384 points
  const int nwg = P / PTS_PER_WG;          // 1024 workgroups

  const size_t smem =
      (size_t)(PTS_PER_WG * PT_STRIDE + PTS_PER_WG * CDIM + PTS_PER_WG * 16 + WAVES) *
      sizeof(float);  // ~271 KB, within CDNA5's 320 KB/workgroup LDS

  (void)hipFuncSetAttribute((const void*)gmm_nll_kernel,
                            hipFuncAttributeMaxDynamicSharedMemorySize, (int)smem);

  gmm_nll_kernel<<<nwg, BLOCK, smem, stream>>>(y, theta, means, covs, partial);
  reduce_kernel<<<1, BLOCK, 0, stream>>>(partial, out, nwg);
}